// AdaptiveNeighborhoodSearch_1606317769120
// MI455X (gfx1250) — compile-verified
//
#include <hip/hip_runtime.h>
#include <math.h>

typedef float v2f __attribute__((ext_vector_type(2)));
typedef float v8f __attribute__((ext_vector_type(8)));

#define KNN_B 2
#define KNN_N 8192
#define KNN_K 16

// One wave (32 threads) per 16-query tile. V_WMMA_F32_16X16X4_F32 computes the
// -2<q,c> 16x16 tile (A pre-scaled by -2, K=3 zero-padded). Lanes q and q+16
// each keep a register-resident sorted top-16 over one contiguous half of the
// candidate columns; a tie-aware LDS merge reproduces jax.top_k ordering.
__global__ __launch_bounds__(32) void knn_wmma_kernel(const float* __restrict__ pts,
                                                      int* __restrict__ out)
{
    const int lane = threadIdx.x;   // 0..31
    const int row  = lane & 15;     // query row / candidate col within tile
    const int hi   = lane >> 4;     // 0: K=0,1 half + cands 0..7; 1: K=2,3 + cands 8..15

    const int ntiles = KNN_N / 16;
    const int b      = blockIdx.x / ntiles;
    const int qbase  = (blockIdx.x % ntiles) * 16;

    const float* P = pts + (size_t)b * KNN_N * 3;

    __shared__ float sA[16];                    // ||q||^2 per query in tile
    __shared__ alignas(16) float tileD[256];    // 16x16 d^2 tile, query-major
    __shared__ float mD[256];                   // merge staging (distances)
    __shared__ int   mI[256];                   // merge staging (indices)

    // ---- A fragment: 16 queries x 4 (K=3 padded with 0), pre-scaled by -2 ----
    float qx = P[(qbase + row) * 3 + 0];
    float qy = P[(qbase + row) * 3 + 1];
    float qz = P[(qbase + row) * 3 + 2];
    if (hi == 0) sA[row] = qx * qx + qy * qy + qz * qz;
    v2f a;
    a.x = hi ? (-2.0f * qz) : (-2.0f * qx);   // K=2 : K=0
    a.y = hi ? 0.0f         : (-2.0f * qy);   // K=3 : K=1
    __syncthreads();

    float sAreg[8];
#pragma unroll
    for (int v = 0; v < 8; ++v) sAreg[v] = sA[v + 8 * hi];

    // per-lane sorted top-16 (ascending distance)
    float bd[KNN_K];
    int   bi[KNN_K];
#pragma unroll
    for (int j = 0; j < KNN_K; ++j) { bd[j] = INFINITY; bi[j] = 0x7fffffff; }

    // ---- software-pipelined candidate coordinate load (hide L2 latency) ----
    float cx = P[row * 3 + 0];
    float cy = P[row * 3 + 1];
    float cz = P[row * 3 + 2];

    for (int cbase = 0; cbase < KNN_N; cbase += 16) {
        // issue next tile's load early; consumed only next iteration
        int nb = (cbase + 16 < KNN_N) ? (cbase + 16) : 0;
        float nx = P[(nb + row) * 3 + 0];
        float ny = P[(nb + row) * 3 + 1];
        float nz = P[(nb + row) * 3 + 2];

        // ---- B fragment: 4 x 16 candidates (K=3 padded), raw coords ----
        float sB = cx * cx + cy * cy + cz * cz;
        v2f bm;
        bm.x = hi ? cz   : cx;
        bm.y = hi ? 0.0f : cy;

        v8f c = {};
        // D = (-2A) x B : 16x16 tile of -2<q,c>
        c = __builtin_amdgcn_wmma_f32_16x16x4_f32(
                /*neg_a=*/false, a, /*neg_b=*/false, bm,
                /*c_mod=*/(short)0, c, /*reuse_a=*/false, /*reuse_b=*/false);

        __syncthreads();  // previous iteration's tileD reads are done
        // lane holds column N=row, rows M = v + 8*hi  ->  transpose via LDS
#pragma unroll
        for (int v = 0; v < 8; ++v) {
            float d2 = c[v] + sAreg[v] + sB;
            tileD[(v + 8 * hi) * 16 + row] = d2;
        }
        __syncthreads();

        // ---- selection: lane (row,hi) scans contiguous candidates ----
        // two b128 LDS loads fetch all 8 distances for this lane's half-row
        const float4* rp = reinterpret_cast<const float4*>(&tileD[row * 16 + hi * 8]);
        float4 dA = rp[0];
        float4 dB = rp[1];
        float dv[8] = { dA.x, dA.y, dA.z, dA.w, dB.x, dB.y, dB.z, dB.w };
        const int ib = cbase + hi * 8;
#pragma unroll
        for (int t = 0; t < 8; ++t) {
            float d   = dv[t];
            int   idx = ib + t;
            if (d < bd[KNN_K - 1]) {        // prune: rarely taken after warm-up
#pragma unroll
                for (int j = 0; j < KNN_K; ++j) {
                    float td = bd[j]; int ti = bi[j];
                    bool  cc = d < td;      // strict < keeps earlier index first
                    bd[j] = cc ? d   : td;
                    bi[j] = cc ? idx : ti;
                    d     = cc ? td  : d;
                    idx   = cc ? ti  : idx;
                }
            }
        }

        cx = nx; cy = ny; cz = nz;
    }

    // ---- merge hi-lane lists into low-lane lists (tie: lower index wins) ----
#pragma unroll
    for (int j = 0; j < KNN_K; ++j) {
        if (hi) { mD[row * 16 + j] = bd[j]; mI[row * 16 + j] = bi[j]; }
    }
    __syncthreads();
    if (hi == 0) {
#pragma unroll
        for (int j = 0; j < KNN_K; ++j) {
            float d   = mD[row * 16 + j];
            int   idx = mI[row * 16 + j];
#pragma unroll
            for (int m = 0; m < KNN_K; ++m) {
                float td = bd[m]; int ti = bi[m];
                bool  cc = (d < td) || ((d == td) && (idx < ti));
                bd[m] = cc ? d   : td;
                bi[m] = cc ? idx : ti;
                d     = cc ? td  : d;
                idx   = cc ? ti  : idx;
            }
        }
        int* o = out + ((size_t)b * KNN_N + qbase + row) * KNN_K;
#pragma unroll
        for (int j = 0; j < KNN_K; ++j) o[j] = bi[j];
    }
}

extern "C" void kernel_launch(void* const* d_in, const int* in_sizes, int n_in,
                              void* d_out, int out_size, void* d_ws, size_t ws_size,
                              hipStream_t stream) {
    (void)in_sizes; (void)n_in; (void)d_ws; (void)ws_size; (void)out_size;
    const float* pts = (const float*)d_in[0];   // (B, N, 3) f32
    int* out = (int*)d_out;                     // (B, N, 16) int32
    dim3 grid(KNN_B * (KNN_N / 16));
    dim3 block(32);
    knn_wmma_kernel<<<grid, block, 0, stream>>>(pts, out);
}